// GATJK_4501125726320
// MI455X (gfx1250) — compile-verified
//
#include <hip/hip_runtime.h>
#include <hip/hip_bf16.h>
#include <math.h>

// ---------------------------------------------------------------------------
// GAT (2x GATConv, heads=2, hid=32) + BN/ELU + JumpingKnowledge(max) + Linear
// MI455X / gfx1250: fp32 WMMA (v_wmma_f32_16x16x4_f32) with LDS-staged B for
// all dense GEMMs; float atomics for the edge-softmax segment reductions.
// ---------------------------------------------------------------------------

#define Nn    100000
#define FIN   128
#define HCc   64      // heads*hid
#define HIDc  32
#define OUTc  40

typedef __attribute__((ext_vector_type(2))) float v2f;
typedef __attribute__((ext_vector_type(8))) float v8f;

// ---------------- float atomic max via order-preserving int punning --------
__device__ inline void atomicMaxFloat(float* addr, float val) {
    int bits = __float_as_int(val);
    if (bits >= 0) {
        atomicMax((int*)addr, bits);                        // +0.0, positives
    } else {
        atomicMin((unsigned int*)addr, (unsigned int)bits); // negatives, -0.0
    }
}

// src/dst for edge e, with implicit self-loops appended after the E real edges
__device__ inline void edge_sd(const int* __restrict__ ei, int e, int E,
                               int& s, int& d) {
    if (e < E) { s = ei[e]; d = ei[E + e]; }
    else       { s = e - E; d = e - E; }
}

// ---------------------------------------------------------------------------
// WMMA GEMM: C[M x Ncols] = Aeff[M x K] @ B[K x Ncols] (+ bias[col])
//   Aeff = A                          (FUSE == false)
//   Aeff = max(A, A2 + a2bias[k])     (FUSE == true; JK-max, final layer)
//
// B is staged once per block into LDS as Bs[K/2][64][2]:
//   Bs[p][col][0..1] = B[2p][col], B[2p+1][col]   (zero-filled past Ncols)
// so the per-iteration B fragment is one 8-byte ds_load at a linear address.
// One wave per 16-row tile; NT (compile-time) 16-col tiles reuse the A frag.
// EXEC stays all-ones across every v_wmma (ISA 7.12 requirement).
// ---------------------------------------------------------------------------
template<int NT, bool FUSE, int K>
__global__ __launch_bounds__(128)
void wmma_gemm(const float* __restrict__ A, const float* __restrict__ A2,
               const float* __restrict__ a2bias,
               const float* __restrict__ B, const float* __restrict__ bias,
               float* __restrict__ C, int M, int Ncols) {
    __shared__ float Bs[(K / 2) * 64 * 2];   // K=128 -> 32KB, K=64 -> 16KB

    // ---- cooperative B staging (guard folded into zero-fill) ----
    for (int i = threadIdx.x; i < K * 64; i += 128) {
        const int p    = i >> 7;        // k-pair index
        const int rem  = i & 127;
        const int col  = rem >> 1;
        const int half = rem & 1;
        Bs[i] = (col < Ncols) ? B[(size_t)(2 * p + half) * Ncols + col] : 0.f;
    }
    __syncthreads();                     // all waves reach this, then may exit

    const int wave = threadIdx.x >> 5;
    const int lane = threadIdx.x & 31;
    const int tileM = blockIdx.x * 4 + wave;
    const int row16 = tileM * 16;
    if (row16 >= M) return;              // wave-uniform exit (after barrier)

    const int hi = (lane >= 16);         // lane-half selects K pair
    const int lr = lane & 15;

    v8f acc[NT];
#pragma unroll
    for (int t = 0; t < NT; ++t)
        acc[t] = (v8f){0.f, 0.f, 0.f, 0.f, 0.f, 0.f, 0.f, 0.f};

    const int arow = row16 + lr;
    const float* __restrict__ Arow  = A + (size_t)arow * K;
    const float* __restrict__ A2row = FUSE ? (A2 + (size_t)arow * K) : nullptr;

#pragma unroll 4
    for (int kb = 0; kb < K; kb += 4) {
        const int ka = kb + 2 * hi;
        // ---- A fragment (16x4 f32): lane holds Aeff[arow][ka], [ka+1]
        float a0 = Arow[ka];
        float a1 = Arow[ka + 1];
        if (FUSE) {
            a0 = fmaxf(a0, A2row[ka]     + a2bias[ka]);
            a1 = fmaxf(a1, A2row[ka + 1] + a2bias[ka + 1]);
        }
        v2f afrag; afrag.x = a0; afrag.y = a1;

        const int p = (kb >> 1) + hi;    // pair row in Bs
#pragma unroll
        for (int t = 0; t < NT; ++t) {
            // contiguous (k,k+1) pair -> single ds_load_b64
            const v2f bfrag = *(const v2f*)&Bs[(p << 7) + ((t * 16 + lr) << 1)];
            acc[t] = __builtin_amdgcn_wmma_f32_16x16x4_f32(
                false, afrag, false, bfrag, (short)0, acc[t], false, false);
        }
    }

    // ---- store D: VGPR r -> rows row16+r (lanes 0-15) / row16+8+r (16-31)
    const int rbase = row16 + (hi ? 8 : 0);
#pragma unroll
    for (int t = 0; t < NT; ++t) {
        const int col = t * 16 + lr;
        if (col < Ncols) {
            const float bb = bias ? bias[col] : 0.f;
#pragma unroll
            for (int r = 0; r < 8; ++r)
                C[(size_t)(rbase + r) * Ncols + col] = acc[t][r] + bb;
        }
    }
}

// ---------------------------------------------------------------------------
// alpha_src/alpha_dst: per (node, head) wave reduction of h . a  (32 lanes)
// ---------------------------------------------------------------------------
__global__ __launch_bounds__(256)
void node_alpha(const float* __restrict__ h, const float* __restrict__ a_s,
                const float* __restrict__ a_d, float* __restrict__ asrc,
                float* __restrict__ adst) {
    const int gw   = (blockIdx.x * blockDim.x + threadIdx.x) >> 5;
    const int lane = threadIdx.x & 31;
    const int n = gw >> 1, hd = gw & 1;
    if (n >= Nn) return;
    const float hv = h[n * HCc + hd * HIDc + lane];
    float vs = hv * a_s[hd * HIDc + lane];
    float vd = hv * a_d[hd * HIDc + lane];
#pragma unroll
    for (int off = 16; off; off >>= 1) {
        vs += __shfl_xor(vs, off, 32);
        vd += __shfl_xor(vd, off, 32);
    }
    if (lane == 0) { asrc[n * 2 + hd] = vs; adst[n * 2 + hd] = vd; }
}

// ---------------------------------------------------------------------------
__global__ __launch_bounds__(256)
void init_layer(float* __restrict__ agg, float* __restrict__ m,
                float* __restrict__ den) {
    const int tid = blockIdx.x * blockDim.x + threadIdx.x;
    if (tid < Nn * HCc) agg[tid] = 0.f;
    if (tid < Nn * 2) { m[tid] = -INFINITY; den[tid] = 0.f; }
}

// pass 1: e = leaky_relu(asrc[s]+adst[d]); segment-max over dst
__global__ __launch_bounds__(256)
void edge_attn_max(const int* __restrict__ ei, int E, int Etot,
                   const float* __restrict__ asrc, const float* __restrict__ adst,
                   float* __restrict__ ebuf, float* __restrict__ m) {
    const int e = blockIdx.x * blockDim.x + threadIdx.x;
    if (e >= Etot) return;
    int s, d; edge_sd(ei, e, E, s, d);
#pragma unroll
    for (int hd = 0; hd < 2; ++hd) {
        float v = asrc[s * 2 + hd] + adst[d * 2 + hd];
        v = (v > 0.f) ? v : 0.2f * v;           // leaky_relu slope 0.2
        ebuf[e * 2 + hd] = v;
        atomicMaxFloat(&m[d * 2 + hd], v);
    }
}

// pass 2: ex = exp(e - m[dst]); segment-sum denom
__global__ __launch_bounds__(256)
void edge_exp_sum(const int* __restrict__ ei, int E, int Etot,
                  const float* __restrict__ m, float* __restrict__ ebuf,
                  float* __restrict__ den) {
    const int e = blockIdx.x * blockDim.x + threadIdx.x;
    if (e >= Etot) return;
    int s, d; edge_sd(ei, e, E, s, d);
#pragma unroll
    for (int hd = 0; hd < 2; ++hd) {
        const float ex = __expf(ebuf[e * 2 + hd] - m[d * 2 + hd]);
        ebuf[e * 2 + hd] = ex;
        atomicAdd(&den[d * 2 + hd], ex);
    }
}

// pass 3: agg[dst] += h[src] * (ex / (den[dst]+1e-16)); 64 lanes per edge
__global__ __launch_bounds__(256)
void edge_msg(const int* __restrict__ ei, int E, int Etot,
              const float* __restrict__ h, const float* __restrict__ ex,
              const float* __restrict__ den, float* __restrict__ agg) {
    const int tid = blockIdx.x * blockDim.x + threadIdx.x;
    if (tid >= Etot * HCc) return;
    const int e = tid >> 6, c = tid & 63, hd = c >> 5;
    int s, d; edge_sd(ei, e, E, s, d);
    const float alpha = ex[e * 2 + hd] / (den[d * 2 + hd] + 1e-16f);
    atomicAdd(&agg[d * HCc + c], h[s * HCc + c] * alpha);
}

// x1 = ELU(BN(agg + b1)), in place
__global__ __launch_bounds__(256)
void node_post1(float* __restrict__ x, const float* __restrict__ b1,
                const float* __restrict__ gamma, const float* __restrict__ beta,
                const float* __restrict__ mean, const float* __restrict__ var) {
    const int tid = blockIdx.x * blockDim.x + threadIdx.x;
    if (tid >= Nn * HCc) return;
    const int c = tid & 63;
    float v = x[tid] + b1[c];
    v = (v - mean[c]) * (gamma[c] * rsqrtf(var[c] + 1e-5f)) + beta[c];
    x[tid] = (v > 0.f) ? v : (__expf(v) - 1.f);  // ELU(alpha=1)
}

// ---------------------------------------------------------------------------
extern "C" void kernel_launch(void* const* d_in, const int* in_sizes, int n_in,
                              void* d_out, int out_size, void* d_ws, size_t ws_size,
                              hipStream_t stream) {
    const float* node_feat = (const float*)d_in[0];
    const int*   ei        = (const int*)  d_in[1];
    const float* W1        = (const float*)d_in[2];
    const float* a_src1    = (const float*)d_in[3];
    const float* a_dst1    = (const float*)d_in[4];
    const float* b1        = (const float*)d_in[5];
    const float* bn_gamma  = (const float*)d_in[6];
    const float* bn_beta   = (const float*)d_in[7];
    const float* bn_mean   = (const float*)d_in[8];
    const float* bn_var    = (const float*)d_in[9];
    const float* W2        = (const float*)d_in[10];
    const float* a_src2    = (const float*)d_in[11];
    const float* a_dst2    = (const float*)d_in[12];
    const float* b2        = (const float*)d_in[13];
    const float* Wf        = (const float*)d_in[14];
    const float* bf        = (const float*)d_in[15];
    float* out = (float*)d_out;

    const int E    = in_sizes[1] / 2;
    const int Etot = E + Nn;

    // ---- workspace carve-out (floats) ----
    float* ws   = (float*)d_ws;
    float* H    = ws;                      // [N,64] h (layer 1 then layer 2)
    float* X1   = H    + (size_t)Nn * HCc; // [N,64] agg1 -> x1 (in place)
    float* AGG2 = X1   + (size_t)Nn * HCc; // [N,64] agg2
    float* ASRC = AGG2 + (size_t)Nn * HCc; // [N,2]
    float* ADST = ASRC + (size_t)Nn * 2;
    float* Mx   = ADST + (size_t)Nn * 2;
    float* DEN  = Mx   + (size_t)Nn * 2;
    float* EX   = DEN  + (size_t)Nn * 2;   // [Etot,2] e -> exp(e-m)

    const int tilesM    = (Nn + 15) / 16;            // 6250 (exact)
    const int gemmBlks  = (tilesM + 3) / 4;
    const int nodeBlks  = (Nn * HCc + 255) / 256;
    const int alphaBlks = (Nn * 2 * 32 + 255) / 256; // wave per (node,head)
    const int edgeBlks  = (Etot + 255) / 256;
    const int msgBlks   = (Etot * HCc + 255) / 256;

    // ================= layer 1 =================
    // h1 = node_feat @ W1   (M=N, K=128, Ncols=64)
    wmma_gemm<4, false, FIN><<<gemmBlks, 128, 0, stream>>>(
        node_feat, nullptr, nullptr, W1, nullptr, H, Nn, HCc);
    node_alpha<<<alphaBlks, 256, 0, stream>>>(H, a_src1, a_dst1, ASRC, ADST);
    init_layer<<<nodeBlks, 256, 0, stream>>>(X1, Mx, DEN);
    edge_attn_max<<<edgeBlks, 256, 0, stream>>>(ei, E, Etot, ASRC, ADST, EX, Mx);
    edge_exp_sum<<<edgeBlks, 256, 0, stream>>>(ei, E, Etot, Mx, EX, DEN);
    edge_msg<<<msgBlks, 256, 0, stream>>>(ei, E, Etot, H, EX, DEN, X1);
    node_post1<<<nodeBlks, 256, 0, stream>>>(X1, b1, bn_gamma, bn_beta,
                                             bn_mean, bn_var);

    // ================= layer 2 =================
    // h2 = x1 @ W2   (K=64)
    wmma_gemm<4, false, HCc><<<gemmBlks, 128, 0, stream>>>(
        X1, nullptr, nullptr, W2, nullptr, H, Nn, HCc);
    node_alpha<<<alphaBlks, 256, 0, stream>>>(H, a_src2, a_dst2, ASRC, ADST);
    init_layer<<<nodeBlks, 256, 0, stream>>>(AGG2, Mx, DEN);
    edge_attn_max<<<edgeBlks, 256, 0, stream>>>(ei, E, Etot, ASRC, ADST, EX, Mx);
    edge_exp_sum<<<edgeBlks, 256, 0, stream>>>(ei, E, Etot, Mx, EX, DEN);
    edge_msg<<<msgBlks, 256, 0, stream>>>(ei, E, Etot, H, EX, DEN, AGG2);

    // ===== final: out = max(x1, agg2 + b2) @ Wf + bf  (K=64, Ncols=40) =====
    wmma_gemm<3, true, HCc><<<gemmBlks, 128, 0, stream>>>(
        X1, AGG2, b2, Wf, bf, out, Nn, OUTc);
}